// CompressorExpander_37211596653106
// MI455X (gfx1250) — compile-verified
//
#include <hip/hip_runtime.h>
#include <math.h>

// ---------------- problem constants (from reference) ----------------
#define BATCH     32
#define TLEN      480000
#define SR        44100.0f
#define LEFT_PAD  44
#define FLEN      706          // sinc filter taps
#define KDIM      724          // Toeplitz K = 721 rounded up to multiple of 4
#define NCHK      (KDIM / 4)   // 181 K-chunks of 4 for f32 WMMA
#define CHUNK     4096         // time chunk per wave in recurrence kernel
#define WARM      16384        // warm-up (state forgets: (1-c)^16384 ~ 1e-7 at 50ms)
#define NCHUNKS   ((TLEN + CHUNK - 1) / CHUNK)   // 118
#define NTILES    (TLEN / 16)  // 30000 time tiles
#define WPB       8            // waves (tiles) per block in WMMA kernel
#define L2_10_O20 0.16609640474f   // log2(10)/20

typedef __attribute__((ext_vector_type(2))) float v2f;
typedef __attribute__((ext_vector_type(8))) float v8f;

// ------------------------------------------------------------------
// Kernel A: build the lane-swizzled Toeplitz filter matrix for WMMA.
// A[i][k] = sinc(k - i - LEFT_PAD - la) for 0 <= k-i <= 705, else 0.
// Swizzled so lane L of any wave loads its (a0,a1) pair for K-chunk c
// as one coalesced float2:  aswz[c*32 + L] = { A[m][4c+koff], A[m][4c+koff+1] }
// with m = L&15, koff = (L>>4)*2   (matches ISA 32-bit A 16x4 layout).
// ------------------------------------------------------------------
__global__ void k_build_A(const float* __restrict__ p_la, v2f* __restrict__ aswz) {
  int idx = blockIdx.x * blockDim.x + threadIdx.x;
  if (idx >= NCHK * 32) return;
  int c = idx >> 5, L = idx & 31;
  int m = L & 15, koff = (L >> 4) * 2;
  float la = p_la[0] * 0.001f * SR;
  v2f v;
#pragma unroll
  for (int q = 0; q < 2; ++q) {
    int k = 4 * c + koff + q;
    int d = k - m;                 // filter tap index
    float val = 0.0f;
    if (d >= 0 && d <= FLEN - 1) {
      float z = (float)d - (float)LEFT_PAD - la;     // sinc argument
      // sin(pi*z) via periodic reduction to [-1,1] so __sinf stays accurate
      float zr = z - 2.0f * rintf(0.5f * z);
      float s  = __sinf(3.14159265358979f * zr);
      val = (fabsf(z) < 1e-6f) ? 1.0f : s / (3.14159265358979f * z);
    }
    v[q] = val;
  }
  aswz[idx] = v;
}

// ------------------------------------------------------------------
// Kernel B: fused envelope + static curve + attack/release.
// Chunk-parallel over time with exponential-forgetting warm-up.
// One wave per chunk, lane = batch row. Writes gain time-major [t][32]
// (coalesced 128B per step, and exactly the B-operand layout the WMMA
// kernel wants).
// ------------------------------------------------------------------
__global__ __launch_bounds__(32)
void k_recur(const float* __restrict__ x,
             const float* __restrict__ p_at, const float* __restrict__ p_rt,
             const float* __restrict__ p_avg,
             const float* __restrict__ p_cr, const float* __restrict__ p_er,
             const float* __restrict__ p_cth, const float* __restrict__ p_eth,
             float* __restrict__ gain_t) {
  const int b  = threadIdx.x;            // lane == batch row (wave32)
  const int t0 = blockIdx.x * CHUNK;
  const int tend = min(TLEN, t0 + CHUNK);
  const int ts = max(0, t0 - WARM);      // ts==0 chunk starts from the exact init

  const float at = p_at[0], rt = p_rt[0], avg = p_avg[0];
  const float cs = 1.0f - 1.0f / p_cr[0];
  const float es = 1.0f - 1.0f / p_er[0];
  const float cth = p_cth[0], eth = p_eth[0];

  const float* xr = x + (long)b * TLEN;
  float lvl = fabsf(xr[ts]);             // exact for chunk 0; converges in warm-up else
  float y   = 1.0f;                      // exact for chunk 0; converges in warm-up else

  for (int t = ts; t < tend; t += 4) {   // ts,tend,CHUNK all multiples of 4 -> aligned
    float4 xv = *(const float4*)(xr + t);
#pragma unroll
    for (int q = 0; q < 4; ++q) {
      float a = fabsf(q == 0 ? xv.x : q == 1 ? xv.y : q == 2 ? xv.z : xv.w);
      lvl += avg * (a - lvl);                                 // one-pole envelope
      float ldb = 20.0f * __log10f(lvl + 1e-10f);
      float gdb = fminf(fminf(cs * (cth - ldb), es * (eth - ldb)), 0.0f);
      float f   = exp2f(gdb * L2_10_O20);                     // 10^(gdb/20)
      float coef = (f < y) ? at : rt;                         // attack when falling
      y += coef * (f - y);
      int tt = t + q;
      if (tt >= t0) gain_t[tt * BATCH + b] = y;
    }
  }
}

// ------------------------------------------------------------------
// Kernel C: 706-tap FIR as Toeplitz matmul via V_WMMA_F32_16X16X4_F32,
// fused with the final out = x * gain_la * makeup.
// D tile: M = 16 consecutive output times, N = 16 batch rows
// (blockIdx.y selects batch half). K = 724, 181 WMMAs per tile.
// Interior tiles (29955 of 30000) take an unclamped fast path so the
// B-operand loads become base-pointer + immediate-offset (no med3/adds).
// ------------------------------------------------------------------
__global__ __launch_bounds__(256)
void k_fir_wmma(const float* __restrict__ x,
                const float* __restrict__ gain_t,
                const v2f*  __restrict__ aswz,
                const float* __restrict__ p_mk,
                float* __restrict__ out) {
  const int lane = threadIdx.x & 31;
  const int wave = threadIdx.x >> 5;
  const int tile = blockIdx.x * WPB + wave;   // < 30000
  const int t0   = tile * 16;
  const int jb   = blockIdx.y * 16;           // batch half
  const int n    = lane & 15;                 // N index (batch col)
  const int koff = (lane >> 4) * 2;           // K sub-slot per ISA layout

  const float* gcol = gain_t + jb + n;        // column of gain for this lane
  const v2f*   arow = aswz + lane;            // this lane's swizzled A stream

  v8f acc = {};
  const int srow0 = t0 - LEFT_PAD + koff;     // first K row (koff-adjusted)

  // Rows touched: [t0-44, t0-44+723]. Clamp needed only for edge tiles.
  if (t0 >= LEFT_PAD && t0 - LEFT_PAD + KDIM - 1 <= TLEN - 1) {
    // ---- fast path: no clamping; compiler strength-reduces to imm offsets
    const float* gptr = gcol + (long)srow0 * BATCH;
#pragma unroll 4
    for (int c = 0; c < NCHK; ++c) {
      v2f a = arow[c * 32];                   // coalesced 256B/wave
      v2f bv;
      bv.x = gptr[(4 * c) * BATCH];
      bv.y = gptr[(4 * c + 1) * BATCH];
      acc = __builtin_amdgcn_wmma_f32_16x16x4_f32(
          false, a, false, bv, (short)0, acc, false, false);
    }
  } else {
    // ---- edge path: replicate padding via clamped row index
    for (int c = 0; c < NCHK; ++c) {
      v2f a = arow[c * 32];
      int r0 = srow0 + 4 * c;
      int r1 = r0 + 1;
      r0 = min(max(r0, 0), TLEN - 1);
      r1 = min(max(r1, 0), TLEN - 1);
      v2f bv;
      bv.x = gcol[r0 * BATCH];
      bv.y = gcol[r1 * BATCH];
      acc = __builtin_amdgcn_wmma_f32_16x16x4_f32(
          false, a, false, bv, (short)0, acc, false, false);
    }
  }

  // Epilogue: lane holds D[m][n] with m = r + 8*(lane>>4)  ->  t = t0 + 8*hi + r,
  // so 8 consecutive times per lane -> two float4 transactions per lane.
  const float mk = exp2f(p_mk[0] * L2_10_O20);
  const int hi = lane >> 4;
  const long base = (long)(jb + n) * TLEN + t0 + hi * 8;
  float4 x0 = *(const float4*)(x + base);
  float4 x1 = *(const float4*)(x + base + 4);
  float4 o0, o1;
  o0.x = x0.x * acc[0] * mk;  o0.y = x0.y * acc[1] * mk;
  o0.z = x0.z * acc[2] * mk;  o0.w = x0.w * acc[3] * mk;
  o1.x = x1.x * acc[4] * mk;  o1.y = x1.y * acc[5] * mk;
  o1.z = x1.z * acc[6] * mk;  o1.w = x1.w * acc[7] * mk;
  *(float4*)(out + base)     = o0;
  *(float4*)(out + base + 4) = o1;
}

// ------------------------------------------------------------------
extern "C" void kernel_launch(void* const* d_in, const int* in_sizes, int n_in,
                              void* d_out, int out_size, void* d_ws, size_t ws_size,
                              hipStream_t stream) {
  const float* x   = (const float*)d_in[0];
  const float* at  = (const float*)d_in[1];
  const float* rt  = (const float*)d_in[2];
  const float* avg = (const float*)d_in[3];
  const float* cr  = (const float*)d_in[4];
  const float* er  = (const float*)d_in[5];
  const float* cth = (const float*)d_in[6];
  const float* eth = (const float*)d_in[7];
  const float* mk  = (const float*)d_in[8];
  const float* la  = (const float*)d_in[9];

  char* ws = (char*)d_ws;
  v2f*   aswz   = (v2f*)ws;                         // 181*32*8 = 46336 B (256-aligned)
  float* gain_t = (float*)(ws + NCHK * 32 * 8);     // TLEN*32*4 = 61.44 MB

  k_build_A<<<dim3((NCHK * 32 + 255) / 256), dim3(256), 0, stream>>>(la, aswz);
  k_recur<<<dim3(NCHUNKS), dim3(32), 0, stream>>>(x, at, rt, avg, cr, er, cth, eth, gain_t);
  k_fir_wmma<<<dim3(NTILES / WPB, 2), dim3(256), 0, stream>>>(x, gain_t, aswz, mk, (float*)d_out);
}